// yolov2_loss_40218073760179
// MI455X (gfx1250) — compile-verified
//
#include <hip/hip_runtime.h>
#include <math.h>

// ---------------------------------------------------------------------------
// YOLOv2 loss on MI455X (gfx1250, wave32).
//   Kernel 1: per-anchor decode + pos-conf/cls/coord losses + compaction of
//             positive GT boxes (only ~2% of 6760 are positive).
//   Kernel 2: best-IoU over compacted positives only (O(N * nPos) instead of
//             O(N^2)) -> negative-conf loss.
//   Kernel 3: single-wave final reduction using V_WMMA_F32_16X16X4_F32 with
//             B = ones (D row-sums == total sum; placement-invariant).
// ---------------------------------------------------------------------------

#define B_ 8
#define H_ 13
#define W_ 13
#define A_ 5
#define C_ 20
#define NTOT (B_ * H_ * W_ * A_)   // 6760
#define NB 27                      // ceil(6760 / 256)

#define L_COORD   3.0f
#define OBJ_SCALE 5.0f

// workspace layout (float units)
#define CNT_OFF  0                       // int counter (16B reserved)
#define PRED_OFF 4                       // 5 arrays of NTOT: x1,y1,x2,y2,area
#define POS_OFF  (PRED_OFF + 5 * NTOT)   // 5 arrays of NTOT: compacted GT boxes
#define P1_OFF   (POS_OFF + 5 * NTOT)    // NB partial sums (pos-dependent losses)
#define P2_OFF   (P1_OFF + 32)           // NB partial sums (neg-conf loss)

__constant__ float c_anchors[2 * A_] = {
    1.3221f,  1.73145f,
    3.19275f, 4.00944f,
    5.05587f, 8.09892f,
    9.47112f, 4.84053f,
    11.2364f, 10.0071f};

__device__ __forceinline__ float block_sum(float v) {
  __shared__ float sm[256];
  sm[threadIdx.x] = v;
  __syncthreads();
#pragma unroll
  for (int s = 128; s > 0; s >>= 1) {
    if ((int)threadIdx.x < s) sm[threadIdx.x] += sm[threadIdx.x + s];
    __syncthreads();
  }
  return sm[0];
}

__device__ __forceinline__ float huber(float d) {
  float ad = fabsf(d);
  return (ad < 1.0f) ? 0.5f * d * d : ad - 0.5f;
}

// --------------------------- Kernel 1 --------------------------------------
__global__ __launch_bounds__(256) void yolo_main(
    const float* __restrict__ pred_cls, const float* __restrict__ pred_conf,
    const float* __restrict__ pred_bboxes, const float* __restrict__ label_cls,
    const float* __restrict__ label_conf,
    const float* __restrict__ label_bboxes, float* __restrict__ ws) {
  const int idx = blockIdx.x * 256 + threadIdx.x;
  float local = 0.0f;

  float* predX1 = ws + PRED_OFF + 0 * NTOT;
  float* predY1 = ws + PRED_OFF + 1 * NTOT;
  float* predX2 = ws + PRED_OFF + 2 * NTOT;
  float* predY2 = ws + PRED_OFF + 3 * NTOT;
  float* predAr = ws + PRED_OFF + 4 * NTOT;
  float* posX1 = ws + POS_OFF + 0 * NTOT;
  float* posY1 = ws + POS_OFF + 1 * NTOT;
  float* posX2 = ws + POS_OFF + 2 * NTOT;
  float* posY2 = ws + POS_OFF + 3 * NTOT;
  float* posAr = ws + POS_OFF + 4 * NTOT;

  if (idx < NTOT) {
    const int a = idx % A_;
    const int w = (idx / A_) % W_;
    const int h = (idx / (A_ * W_)) % H_;
    const float ancx = (float)w, ancy = (float)h;
    const float ancw = c_anchors[2 * a], anch = c_anchors[2 * a + 1];
    const float INV = 1.0f / 13.0f;  // fs0 == fs1 == 13

    // decode pred box
    const float* pb = pred_bboxes + (size_t)idx * 4;
    float pw = expf(pb[2]) * ancw * INV;
    float ph = expf(pb[3]) * anch * INV;
    float px = (pb[0] + ancx) * INV - 0.5f * pw;
    float py = (pb[1] + ancy) * INV - 0.5f * ph;
    predX1[idx] = px;
    predY1[idx] = py;
    predX2[idx] = px + pw;
    predY2[idx] = py + ph;
    predAr[idx] = pw * ph;

    const bool pos = label_conf[idx] > 0.0f;
    if (pos) {
      // decode gt box
      const float* lb = label_bboxes + (size_t)idx * 4;
      float gw = expf(lb[2]) * ancw * INV;
      float gh = expf(lb[3]) * anch * INV;
      float gx = (lb[0] + ancx) * INV - 0.5f * gw;
      float gy = (lb[1] + ancy) * INV - 0.5f * gh;

      // compact positive GT boxes (order-independent downstream: fed to max)
      int slot = atomicAdd((int*)ws, 1);
      posX1[slot] = gx;
      posY1[slot] = gy;
      posX2[slot] = gx + gw;
      posY2[slot] = gy + gh;
      posAr[slot] = gw * gh;

      // diagonal IoU for positive-conf loss
      float dx = fminf(px + pw, gx + gw) - fmaxf(px, gx);
      float dy = fminf(py + ph, gy + gh) - fmaxf(py, gy);
      dx = fmaxf(dx, 0.0f);
      dy = fmaxf(dy, 0.0f);
      float inter = dx * dy;
      float iou = inter / (gw * gh + pw * ph - inter);
      float dconf = pred_conf[idx] - iou;
      local += OBJ_SCALE * dconf * dconf;

      // cls NLL: lbl = argmax(label_cls) (first max wins), nll = lse - x[lbl]
      const float* lc = label_cls + (size_t)idx * C_;
      const float* pc = pred_cls + (size_t)idx * C_;
      float bestv = lc[0];
      int lbl = 0;
      float xmax = pc[0];
#pragma unroll
      for (int c = 1; c < C_; ++c) {
        float v = lc[c];
        if (v > bestv) { bestv = v; lbl = c; }
        xmax = fmaxf(xmax, pc[c]);
      }
      float sexp = 0.0f;
#pragma unroll
      for (int c = 0; c < C_; ++c) sexp += expf(pc[c] - xmax);
      float lse = xmax + logf(sexp);
      local += lse - pc[lbl];

      // coord losses on raw deltas
      float d0 = pb[0] - lb[0];
      float d1 = pb[1] - lb[1];
      local += L_COORD * (d0 * d0 + d1 * d1 + huber(pb[2] - lb[2]) +
                          huber(pb[3] - lb[3]));
    }
  }

  float s = block_sum(local);
  if (threadIdx.x == 0) ws[P1_OFF + blockIdx.x] = s;
}

// --------------------------- Kernel 2 --------------------------------------
__global__ __launch_bounds__(256) void yolo_neg(
    const float* __restrict__ pred_conf, const float* __restrict__ label_conf,
    float* __restrict__ ws) {
  const int j = blockIdx.x * 256 + threadIdx.x;
  float local = 0.0f;

  const float* predX1 = ws + PRED_OFF + 0 * NTOT;
  const float* predY1 = ws + PRED_OFF + 1 * NTOT;
  const float* predX2 = ws + PRED_OFF + 2 * NTOT;
  const float* predY2 = ws + PRED_OFF + 3 * NTOT;
  const float* predAr = ws + PRED_OFF + 4 * NTOT;
  const float* posX1 = ws + POS_OFF + 0 * NTOT;
  const float* posY1 = ws + POS_OFF + 1 * NTOT;
  const float* posX2 = ws + POS_OFF + 2 * NTOT;
  const float* posY2 = ws + POS_OFF + 3 * NTOT;
  const float* posAr = ws + POS_OFF + 4 * NTOT;

  if (j < NTOT) {
    const int nPos = *(const int*)ws;
    const float lc = label_conf[j];
    float best;
    if (lc > 0.0f) {
      best = 1.0f;  // reference: best = 1 for positive columns
    } else {
      best = 0.0f;  // masked rows contribute exactly 0
      const float x1 = predX1[j], y1 = predY1[j];
      const float x2 = predX2[j], y2 = predY2[j];
      const float ar = predAr[j];
      for (int k = 0; k < nPos; ++k) {  // uniform (scalar-cache-friendly) loads
        float dx = fminf(x2, posX2[k]) - fmaxf(x1, posX1[k]);
        float dy = fminf(y2, posY2[k]) - fmaxf(y1, posY1[k]);
        dx = fmaxf(dx, 0.0f);
        dy = fmaxf(dy, 0.0f);
        float inter = dx * dy;
        best = fmaxf(best, inter / (ar + posAr[k] - inter));
      }
    }
    if (best < 0.6f) {
      float d = pred_conf[j] - lc;
      local = d * d;  // NOOBJ_SCALE = 1
    }
  }

  float s = block_sum(local);
  if (threadIdx.x == 0) ws[P2_OFF + blockIdx.x] = s;
}

// --------------------------- Kernel 3 (WMMA reduction) ---------------------
typedef __attribute__((ext_vector_type(2))) float v2f;
typedef __attribute__((ext_vector_type(8))) float v8f;

__global__ __launch_bounds__(32) void yolo_final(const float* __restrict__ ws,
                                                 float* __restrict__ out) {
  const int lane = threadIdx.x;
  // 54 partials (27 + 27), zero-padded to the 64 slots of a 16x4 f32 A-matrix.
  // B = ones => D[m][n] = sum_k A[m][k]; summing all rows gives the total sum
  // independent of which (m,k) slot each value occupies (bijective packing).
  auto fetch = [&](int i) -> float {
    float v = 0.0f;
    if (i < NB) v = ws[P1_OFF + i];
    else if (i < 2 * NB) v = ws[P2_OFF + (i - NB)];
    return v;
  };
  v2f a;
  a.x = fetch(2 * lane);
  a.y = fetch(2 * lane + 1);
  v2f b;
  b.x = 1.0f;
  b.y = 1.0f;
  v8f c = {};
  // 8 args: (neg_a, A, neg_b, B, c_mod, C, reuse_a, reuse_b)
  c = __builtin_amdgcn_wmma_f32_16x16x4_f32(false, a, false, b, (short)0, c,
                                            false, false);
  // lanes 0-15 hold rows 0-7 in c[0..7]; lanes 16-31 hold rows 8-15.
  float s = c[0] + c[1] + c[2] + c[3] + c[4] + c[5] + c[6] + c[7];
  float total = __shfl(s, 0, 32) + __shfl(s, 16, 32);
  if (lane == 0) out[0] = total * (1.0f / (float)B_);
}

// --------------------------- Launch ----------------------------------------
extern "C" void kernel_launch(void* const* d_in, const int* in_sizes, int n_in,
                              void* d_out, int out_size, void* d_ws,
                              size_t ws_size, hipStream_t stream) {
  const float* pred_cls = (const float*)d_in[0];
  const float* pred_conf = (const float*)d_in[1];
  const float* pred_bboxes = (const float*)d_in[2];
  const float* label_cls = (const float*)d_in[3];
  const float* label_conf = (const float*)d_in[4];
  const float* label_bboxes = (const float*)d_in[5];
  float* ws = (float*)d_ws;

  hipMemsetAsync(d_ws, 0, 16, stream);  // reset compaction counter
  yolo_main<<<NB, 256, 0, stream>>>(pred_cls, pred_conf, pred_bboxes,
                                    label_cls, label_conf, label_bboxes, ws);
  yolo_neg<<<NB, 256, 0, stream>>>(pred_conf, label_conf, ws);
  yolo_final<<<1, 32, 0, stream>>>(ws, (float*)d_out);
}